// SmallNet_28656021798994
// MI455X (gfx1250) — compile-verified
//
#include <hip/hip_runtime.h>
#include <hip/hip_bf16.h>

#define N_POINTS   1024
#define MC_SAMPLES 64
#define N_EVENTS   100000
#define N_PAIRS    (N_POINTS * (N_POINTS - 1) / 2)   // 523776
#define EPS        1e-06f

#define EVENT_BLOCKS 256
#define PAIR_BLOCKS  1024
#define BLOCK        256
#define WAVES_PER_BLOCK (BLOCK / 32)
#define TOTAL_WAVES  (PAIR_BLOCKS * WAVES_PER_BLOCK)

typedef __attribute__((ext_vector_type(2))) float v2f;
typedef __attribute__((ext_vector_type(2))) int   v2i;
typedef __attribute__((ext_vector_type(8))) float v8f;

// ---------------------------------------------------------------------------
// Wave-wide sum of one float per lane on the matrix pipe (wave32).
// Pass 1: A(16x4) holds lane L's value at A[L%16][0] (lanes 0-15) and
//         A[L%16][2] (lanes 16-31); B = ones(4x16) -> D[m][n] = v_m + v_{m+16}.
//         Row-summing each lane's 8 D regs yields S_low (lanes 0-15) /
//         S_high (lanes 16-31).
// Pass 2: same trick on {S_low, S_high} broadcasts the full wave sum to every
//         lane. EXEC must be all 1s (no divergence in the caller).
// ---------------------------------------------------------------------------
__device__ __forceinline__ float wave_sum_wmma(float v) {
  v2f a; a[0] = v;    a[1] = 0.0f;
  v2f b; b[0] = 1.0f; b[1] = 1.0f;
  v8f c = {};
  v8f d1 = __builtin_amdgcn_wmma_f32_16x16x4_f32(false, a, false, b,
                                                 (short)0, c, false, false);
  float s = ((d1[0] + d1[1]) + (d1[2] + d1[3])) +
            ((d1[4] + d1[5]) + (d1[6] + d1[7]));
  v2f a2; a2[0] = s; a2[1] = 0.0f;
  v8f d2 = __builtin_amdgcn_wmma_f32_16x16x4_f32(false, a2, false, b,
                                                 (short)0, c, false, false);
  return d2[0];
}

// offset of first pair in row i of the k=1 upper triangle
__device__ __forceinline__ int triu_off(int i) {
  return (i * (2 * N_POINTS - 1 - i)) >> 1;
}

// Async global->LDS stage of one 8-byte element per lane (256B per wave).
// ldsoff is the lane's LDS byte offset, gbase the row base pointer.
__device__ __forceinline__ void async_row_load(const float* gbase,
                                               unsigned ldsoff, int lane) {
  unsigned long long ga =
      (unsigned long long)(uintptr_t)gbase + (unsigned long long)(8u * lane);
  asm volatile("global_load_async_to_lds_b64 %0, %1, off"
               :: "v"(ldsoff), "v"(ga)
               : "memory");
}

// ---------------------------------------------------------------------------
// Kernel 1: event intensity partials.  sum_e (beta - ||dz + dv*t + eps||)
// Single-use streams (uv, t) use non-temporal b64/b32 loads; the 16KB
// z0/v0 tables stay cache-resident.
// ---------------------------------------------------------------------------
__global__ __launch_bounds__(BLOCK)
void event_kernel(const float* __restrict__ z0, const float* __restrict__ v0,
                  const float* __restrict__ beta, const int* __restrict__ uv,
                  const float* __restrict__ et, float* __restrict__ partials) {
  const float bv = beta[0];
  const v2i* uv2 = (const v2i*)uv;
  const v2f* z2  = (const v2f*)z0;
  const v2f* v2  = (const v2f*)v0;
  float acc = 0.0f;
  for (int e = blockIdx.x * BLOCK + threadIdx.x; e < N_EVENTS;
       e += gridDim.x * BLOCK) {
    const v2i p  = __builtin_nontemporal_load(&uv2[e]);
    const float t = __builtin_nontemporal_load(&et[e]);
    const v2f zu = z2[p[0]], zv = z2[p[1]];
    const v2f vu = v2[p[0]], vv = v2[p[1]];
    const float dx = fmaf(vu[0] - vv[0], t, zu[0] - zv[0]) + EPS;
    const float dy = fmaf(vu[1] - vv[1], t, zu[1] - zv[1]) + EPS;
    acc += bv - sqrtf(fmaf(dx, dx, dy * dy));
  }
  __shared__ float s[BLOCK];
  s[threadIdx.x] = acc;
  __syncthreads();
  for (int st = BLOCK / 2; st > 0; st >>= 1) {
    if (threadIdx.x < st) s[threadIdx.x] += s[threadIdx.x + st];
    __syncthreads();
  }
  if (threadIdx.x == 0) partials[blockIdx.x] = s[0];
}

// ---------------------------------------------------------------------------
// Kernel 2: non-event MC partials.  One wave per pair per grid-stride step.
// samples_u is streamed via the CDNA5 async engine into a per-wave LDS double
// buffer (global_load_async_to_lds_b64 + s_wait_asynccnt), overlapping HBM
// latency with the trans-pipe work: d = sqrt(a + t*(b + t*c)), lam = exp(b-d).
// Raw lam sum accumulated; uniform scale/64 applied at the end.
// ---------------------------------------------------------------------------
__global__ __launch_bounds__(BLOCK)
void pair_kernel(const float* __restrict__ z0, const float* __restrict__ v0,
                 const float* __restrict__ beta,
                 const float* __restrict__ samples_u,
                 const int* __restrict__ t0p, const int* __restrict__ tnp,
                 float* __restrict__ partials) {
  __shared__ float sbuf[WAVES_PER_BLOCK][2][MC_SAMPLES];  // 4KB double buffer
  __shared__ float wsum[WAVES_PER_BLOCK];

  const int lane = threadIdx.x & 31;
  const int wid  = threadIdx.x >> 5;
  const int wave = blockIdx.x * WAVES_PER_BLOCK + wid;

  const float bv    = beta[0];
  const float tbase = (float)(*t0p);
  const float scale = (float)(*tnp - *t0p);

  // per-lane LDS byte offsets (8B per lane) for the two buffers
  float* buf_cur = &sbuf[wid][0][0];
  float* buf_nxt = &sbuf[wid][1][0];
  unsigned off_cur = (unsigned)(uintptr_t)buf_cur + 8u * (unsigned)lane;
  unsigned off_nxt = (unsigned)(uintptr_t)buf_nxt + 8u * (unsigned)lane;

  // prologue: stage this wave's first row (every wave owns >= 63 pairs)
  async_row_load(samples_u + (size_t)wave * MC_SAMPLES, off_cur, lane);

  float lam_acc = 0.0f;  // per-lane raw sum of exp(beta - d)

  for (int p = wave; p < N_PAIRS; p += TOTAL_WAVES) {
    // stage next row into the other buffer (clamped re-load on last iter)
    const int pn = p + TOTAL_WAVES;
    const float* nxt_row =
        samples_u + (size_t)((pn < N_PAIRS) ? pn : p) * MC_SAMPLES;
    async_row_load(nxt_row, off_nxt, lane);
    // deeper L2 prefetch (global_prefetch_b8)
    if (p + 2 * TOTAL_WAVES < N_PAIRS)
      __builtin_prefetch(
          samples_u + (size_t)(p + 2 * TOTAL_WAVES) * MC_SAMPLES + lane, 0, 1);

    // invert p -> (i, j) for jnp.triu_indices(N, k=1) row-major ordering
    const float disc = (float)((2 * N_POINTS - 1) * (2 * N_POINTS - 1)) -
                       8.0f * (float)p;
    int i = (int)floorf(((float)(2 * N_POINTS - 1) - sqrtf(disc)) * 0.5f);
    while (triu_off(i + 1) <= p) ++i;
    while (triu_off(i) > p) --i;
    const int j = i + 1 + (p - triu_off(i));

    // pair coefficients (lane-uniform, L2-resident 16KB tables)
    const float ax  = z0[2 * i]     - z0[2 * j]     + EPS;
    const float ay  = z0[2 * i + 1] - z0[2 * j + 1] + EPS;
    const float dvx = v0[2 * i]     - v0[2 * j];
    const float dvy = v0[2 * i + 1] - v0[2 * j + 1];
    const float qa = fmaf(ax, ax, ay * ay);
    const float qb = 2.0f * fmaf(ax, dvx, ay * dvy);
    const float qc = fmaf(dvx, dvx, dvy * dvy);

    // wait for current row: 1 outstanding (the just-issued next row) allowed;
    // async loads complete in order, so buf_cur is ready after this.
    asm volatile("s_wait_asynccnt 0x1" ::: "memory");

    const float u0 = buf_cur[2 * lane];
    const float u1 = buf_cur[2 * lane + 1];
    float t, d;
    t = fmaf(scale, u0, tbase);
    d = sqrtf(fmaf(t, fmaf(t, qc, qb), qa));
    lam_acc += __expf(bv - d);
    t = fmaf(scale, u1, tbase);
    d = sqrtf(fmaf(t, fmaf(t, qc, qb), qa));
    lam_acc += __expf(bv - d);

    // swap double buffers
    float* tb = buf_cur; buf_cur = buf_nxt; buf_nxt = tb;
    unsigned to = off_cur; off_cur = off_nxt; off_nxt = to;
  }

  // drain the async engine before exiting
  asm volatile("s_wait_asynccnt 0x0" ::: "memory");

  // wave reduction on the matrix pipe (v_wmma_f32_16x16x4_f32 x2)
  const float wave_total = wave_sum_wmma(lam_acc);

  if (lane == 0) wsum[wid] = wave_total;
  __syncthreads();
  if (threadIdx.x == 0) {
    float bsum = 0.0f;
    for (int k = 0; k < WAVES_PER_BLOCK; ++k) bsum += wsum[k];  // fixed order
    partials[blockIdx.x] = bsum;
  }
}

// ---------------------------------------------------------------------------
// Kernel 3: deterministic final reduction + outputs
//   out[0] = log_likelihood = E - NE,   out[1] = E / NE
//   NE = (tn - t0) * lam_sum / MC_SAMPLES
// ---------------------------------------------------------------------------
__global__ __launch_bounds__(BLOCK)
void finalize_kernel(const float* __restrict__ evp, int nev,
                     const float* __restrict__ prp, int npr,
                     const int* __restrict__ t0p, const int* __restrict__ tnp,
                     float* __restrict__ out) {
  __shared__ float s[BLOCK];
  const int tid = threadIdx.x;

  float a = 0.0f;
  for (int k = tid; k < nev; k += BLOCK) a += evp[k];
  s[tid] = a;
  __syncthreads();
  for (int st = BLOCK / 2; st > 0; st >>= 1) {
    if (tid < st) s[tid] += s[tid + st];
    __syncthreads();
  }
  const float E = s[0];
  __syncthreads();

  float c = 0.0f;
  for (int k = tid; k < npr; k += BLOCK) c += prp[k];
  s[tid] = c;
  __syncthreads();
  for (int st = BLOCK / 2; st > 0; st >>= 1) {
    if (tid < st) s[tid] += s[tid + st];
    __syncthreads();
  }

  if (tid == 0) {
    const float scale = (float)(*tnp - *t0p);
    const float NE = s[0] * scale * (1.0f / (float)MC_SAMPLES);
    out[0] = E - NE;
    out[1] = E / NE;
  }
}

extern "C" void kernel_launch(void* const* d_in, const int* in_sizes, int n_in,
                              void* d_out, int out_size, void* d_ws,
                              size_t ws_size, hipStream_t stream) {
  const float* z0        = (const float*)d_in[0];
  const float* v0        = (const float*)d_in[1];
  const float* beta      = (const float*)d_in[2];
  const int*   event_uv  = (const int*)d_in[3];
  const float* event_t   = (const float*)d_in[4];
  const float* samples_u = (const float*)d_in[5];
  const int*   t0p       = (const int*)d_in[6];
  const int*   tnp       = (const int*)d_in[7];
  float*       out       = (float*)d_out;

  float* ws            = (float*)d_ws;
  float* ev_partials   = ws;                  // EVENT_BLOCKS floats
  float* pair_partials = ws + EVENT_BLOCKS;   // PAIR_BLOCKS floats

  event_kernel<<<EVENT_BLOCKS, BLOCK, 0, stream>>>(z0, v0, beta, event_uv,
                                                   event_t, ev_partials);
  pair_kernel<<<PAIR_BLOCKS, BLOCK, 0, stream>>>(z0, v0, beta, samples_u, t0p,
                                                 tnp, pair_partials);
  finalize_kernel<<<1, BLOCK, 0, stream>>>(ev_partials, EVENT_BLOCKS,
                                           pair_partials, PAIR_BLOCKS, t0p,
                                           tnp, out);
}